// Agent_25778393711265
// MI455X (gfx1250) — compile-verified
//
#include <hip/hip_runtime.h>

typedef float v2f __attribute__((ext_vector_type(2)));
typedef float v4f __attribute__((ext_vector_type(4)));
typedef float v8f __attribute__((ext_vector_type(8)));

#define WPB 4  // waves per block (wave32)

// Map padded feat column k' (0..31) -> original W1 row, or -1 for zero pad.
// Padded feat layout: [0..7]=price_prev, [8]=cash, [9]=pv, [10]=1.0 (bias),
//                     [11..15]=0, [16..23]=pos[0..7], [24..31]=0
// Original feat:      [0..7]=price_prev, [8]=cash, [9..16]=pos, [17]=pv
__device__ __forceinline__ int mapk(int kp) {
  if (kp < 8) return kp;
  if (kp == 8) return 8;
  if (kp == 9) return 17;
  if (kp >= 16 && kp < 24) return kp - 7;
  return -1;
}

__global__ __launch_bounds__(32 * WPB) void hedge_rollout(
    const float* __restrict__ hp,   // (P, T, 8)
    const float* __restrict__ W1,   // (18, 64)
    const float* __restrict__ b1,   // (64,)
    const float* __restrict__ W2,   // (64, 8)
    const float* __restrict__ b2,   // (8,)
    float* __restrict__ out,        // (P,)
    int P, int T)
{
  __shared__ float feat[WPB][16][32];  // padded feature rows (per wave)
  __shared__ float hT[WPB][64][16];    // h transposed: [hidden][path]

  const int lane = threadIdx.x & 31;
  const int wid  = threadIdx.x >> 5;
  const int hi   = lane >> 4;   // half-wave select
  const int jj   = lane & 15;

  // zero feat so pad columns stay zero forever; col 10 = 1.0 (bias carrier)
  for (int i = threadIdx.x; i < WPB * 16 * 32; i += blockDim.x)
    (&feat[0][0][0])[i] = 0.0f;
  __syncthreads();
  if (lane < 16) feat[wid][jj][10] = 1.0f;
  __syncthreads();

  // ---- Build weight fragments in registers (held for the whole rollout) ----
  // Layer 1 B fragments: B 4x16 layout: element(k,n): vgpr=(k&1), lane=n+16*(k>=2)
  const int chk[5] = {0, 1, 2, 4, 5};  // non-zero K-chunks of padded feat
  float b1f[4][5][2];
#pragma unroll
  for (int t4 = 0; t4 < 4; ++t4)
#pragma unroll
    for (int ci = 0; ci < 5; ++ci)
#pragma unroll
      for (int p = 0; p < 2; ++p) {
        int kp = 4 * chk[ci] + 2 * hi + p;
        int ko = mapk(kp);
        float v;
        if (kp == 10)      v = b1[16 * t4 + jj];          // bias row
        else if (ko >= 0)  v = W1[ko * 64 + 16 * t4 + jj];
        else               v = 0.0f;
        b1f[t4][ci][p] = v;
      }

  // Layer 2 A fragments (W2^T): A 16x4 layout: element(m,k): vgpr=(k&1),
  // lane=m+16*(k>=2). Row j = action component (rows 8..15 zero-padded).
  float b2f[16][2];
#pragma unroll
  for (int c2 = 0; c2 < 16; ++c2)
#pragma unroll
    for (int p = 0; p < 2; ++p) {
      int k = 4 * c2 + 2 * hi + p;
      b2f[c2][p] = (jj < 8) ? W2[k * 8 + jj] : 0.0f;
    }
  // Uniform bias values (become scalar regs)
  float b2s[8];
#pragma unroll
  for (int r = 0; r < 8; ++r) b2s[r] = b2[r];

  // ---- Per-lane state: lanes 0..15 each own one path ----
  const int path = blockIdx.x * (WPB * 16) + wid * 16 + jj;
  const bool own = (lane < 16) && (path < P);

  float cash = 0.0f, pv = 0.0f;
  float pos[8], ppv[8], ptv[8];
#pragma unroll
  for (int j = 0; j < 8; ++j) { pos[j] = 0.0f; ppv[j] = 0.0f; ptv[j] = 0.0f; }

  if (own) {
    const float* p0 = hp + (long long)path * T * 8;
    v4f q0 = __builtin_nontemporal_load(reinterpret_cast<const v4f*>(p0));
    v4f q1 = __builtin_nontemporal_load(reinterpret_cast<const v4f*>(p0 + 4));
#pragma unroll
    for (int j = 0; j < 4; ++j) { ppv[j] = q0[j]; ppv[4 + j] = q1[j]; }
  }

  const v8f vzero = {};

  for (int t = 1; t < T; ++t) {
    // -- load this step's prices + write feature row (one divergent region) --
    if (own) {
      const float* pr = hp + ((long long)path * T + t) * 8;
      v4f q0 = __builtin_nontemporal_load(reinterpret_cast<const v4f*>(pr));
      v4f q1 = __builtin_nontemporal_load(reinterpret_cast<const v4f*>(pr + 4));
#pragma unroll
      for (int j = 0; j < 4; ++j) { ptv[j] = q0[j]; ptv[4 + j] = q1[j]; }
      __builtin_prefetch(pr + 8, 0, 0);  // next step's row

      float* fr = &feat[wid][jj][0];
      *reinterpret_cast<float4*>(fr +  0) = make_float4(ppv[0], ppv[1], ppv[2], ppv[3]);
      *reinterpret_cast<float4*>(fr +  4) = make_float4(ppv[4], ppv[5], ppv[6], ppv[7]);
      *reinterpret_cast<v2f*>(fr + 8) = (v2f){cash, pv};
      *reinterpret_cast<float4*>(fr + 16) = make_float4(pos[0], pos[1], pos[2], pos[3]);
      *reinterpret_cast<float4*>(fr + 20) = make_float4(pos[4], pos[5], pos[6], pos[7]);
    }
    __builtin_amdgcn_wave_barrier();

    // -- layer 1: h(16x64) = feat(16x32) @ W1p(32x64), skipping zero chunks --
    v2f a1[5];
#pragma unroll
    for (int ci = 0; ci < 5; ++ci)
      a1[ci] = *reinterpret_cast<const v2f*>(&feat[wid][jj][4 * chk[ci] + 2 * hi]);

    v8f hc[4];
#pragma unroll
    for (int t4 = 0; t4 < 4; ++t4) {
      v2f b0 = {b1f[t4][0][0], b1f[t4][0][1]};
      hc[t4] = __builtin_amdgcn_wmma_f32_16x16x4_f32(
          false, a1[0], false, b0, (short)0, vzero, false, false);
#pragma unroll
      for (int ci = 1; ci < 5; ++ci) {
        v2f b = {b1f[t4][ci][0], b1f[t4][ci][1]};
        hc[t4] = __builtin_amdgcn_wmma_f32_16x16x4_f32(
            false, a1[ci], false, b, (short)0, hc[t4], false, false);
      }
    }

    // ReLU
#pragma unroll
    for (int t4 = 0; t4 < 4; ++t4)
#pragma unroll
      for (int r = 0; r < 8; ++r) hc[t4][r] = fmaxf(hc[t4][r], 0.0f);

    // store h transposed: hT[hidden][path]; contiguous over r -> b128 stores
#pragma unroll
    for (int t4 = 0; t4 < 4; ++t4) {
      float* hr = &hT[wid][16 * t4 + jj][8 * hi];
      *reinterpret_cast<float4*>(hr)     = make_float4(hc[t4][0], hc[t4][1], hc[t4][2], hc[t4][3]);
      *reinterpret_cast<float4*>(hr + 4) = make_float4(hc[t4][4], hc[t4][5], hc[t4][6], hc[t4][7]);
    }
    __builtin_amdgcn_wave_barrier();

    // -- layer 2 (transposed): action^T(16x16) = W2^T(16x64) @ h^T(64x16) --
    // A = b2f (weights), B = h^T chunks loaded from hT (same addresses as before).
    // Result C-layout: row j (action component) in VGPR j, columns (paths) in
    // lanes 0..15 -> action lands directly in the state-owning lanes.
    v8f acc[4];
#pragma unroll
    for (int r = 0; r < 8; ++r) acc[0][r] = (lane < 16) ? b2s[r] : 0.0f;
    acc[1] = vzero; acc[2] = vzero; acc[3] = vzero;
#pragma unroll
    for (int c2 = 0; c2 < 16; ++c2) {
      int row = 4 * c2 + 2 * hi;
      v2f hfrag = {hT[wid][row][jj], hT[wid][row + 1][jj]};
      v2f wfrag = {b2f[c2][0], b2f[c2][1]};
      acc[c2 >> 2] = __builtin_amdgcn_wmma_f32_16x16x4_f32(
          false, wfrag, false, hfrag, (short)0, acc[c2 >> 2], false, false);
    }
    float act[8];
#pragma unroll
    for (int r = 0; r < 8; ++r)
      act[r] = (acc[0][r] + acc[1][r]) + (acc[2][r] + acc[3][r]);

    // -- state update: act[j] is already in the owning lane's registers --
    float dotp = 0.0f, dota = 0.0f, dotn = 0.0f;
#pragma unroll
    for (int j = 0; j < 8; ++j) {
      float np = pos[j] + act[j];
      dotp = fmaf(act[j], ptv[j], dotp);
      dota = fmaf(fabsf(act[j]), ppv[j], dota);
      dotn = fmaf(np, ptv[j], dotn);
      pos[j] = np;
      ppv[j] = ptv[j];
    }
    cash = cash * 1.05f - (dotp + 1e-3f * dota);
    pv = dotn + cash;

    __builtin_amdgcn_wave_barrier();
  }

  if (own) out[path] = pv + cash;
}

extern "C" void kernel_launch(void* const* d_in, const int* in_sizes, int n_in,
                              void* d_out, int out_size, void* d_ws, size_t ws_size,
                              hipStream_t stream) {
  const float* hp = (const float*)d_in[0];
  const float* W1 = (const float*)d_in[1];
  const float* b1 = (const float*)d_in[2];
  const float* W2 = (const float*)d_in[3];
  const float* b2 = (const float*)d_in[4];
  float* out = (float*)d_out;

  int P = out_size;
  int T = in_sizes[0] / (P * 8);

  dim3 block(32 * WPB);
  dim3 grid((P + (WPB * 16) - 1) / (WPB * 16));
  hedge_rollout<<<grid, block, 0, stream>>>(hp, W1, b1, W2, b2, out, P, T);
}